// GcnNet_84112639525116
// MI455X (gfx1250) — compile-verified
//
#include <hip/hip_runtime.h>
#include <hip/hip_bf16.h>

typedef float v8f __attribute__((ext_vector_type(8)));
typedef float v2f __attribute__((ext_vector_type(2)));

#define N_NODES    50000
#define N_EDGES    800000
#define DIM        64
#define NUM_GRAPHS 512
#define FC1        128

static __device__ __forceinline__ void atomic_add_f32(float* p, float v) {
    __hip_atomic_fetch_add(p, v, __ATOMIC_RELAXED, __HIP_MEMORY_SCOPE_AGENT);
}

__global__ void zero_kernel(float* __restrict__ p, int n) {
    int i = blockIdx.x * blockDim.x + threadIdx.x;
    if (i < n) p[i] = 0.0f;
}

__global__ void degree_kernel(const int* __restrict__ dst, float* __restrict__ deg, int E) {
    int e = blockIdx.x * blockDim.x + threadIdx.x;
    if (e < E) atomic_add_f32(&deg[dst[e]], 1.0f);
}

__global__ void rsqrt_kernel(float* __restrict__ d, int n) {
    int i = blockIdx.x * blockDim.x + threadIdx.x;
    if (i < n) d[i] = rsqrtf(d[i] + 1.0f);
}

// C[M,N] = A[M,K] @ B[K,N] (+bias, +relu), row-major, compile-time K/N.
// One wave computes a 16 x (16*NT) strip: NT f32 accumulators, A fragment
// reused across NT WMMAs per k-step, K fully unrolled (K/4 * NT wmma ops).
template <int K, int N, int NT, bool BIAS, bool RELU>
__global__ void wmma_gemm(const float* __restrict__ A, const float* __restrict__ B,
                          const float* __restrict__ bias, float* __restrict__ C,
                          int Mtiles) {
    constexpr int COLG = N / (16 * NT);           // col-groups per row-tile
    int wave = blockIdx.x * (blockDim.x >> 5) + (threadIdx.x >> 5);
    int tm = wave / COLG;
    int tc = wave % COLG;
    if (tm >= Mtiles) return;                     // wave-uniform: EXEC all-1s inside

    int lane = threadIdx.x & 31;
    int l    = lane & 15;                         // row (A) / col (B,C)
    int hi   = lane >> 4;                         // K-pair / row-half select
    int row0 = tm << 4;
    int n0   = tc * (16 * NT);

    v8f acc[NT] = {};

    const float* arow = A + (row0 + l) * K + 2 * hi;
    const float* bcol = B + n0 + l + 2 * hi * N;
#pragma unroll
    for (int k = 0; k < K; k += 4) {
        // A fragment: VGPR0 = A[row][k+2*hi], VGPR1 = A[row][k+2*hi+1]
        v2f a = *(const v2f*)(arow + k);
#pragma unroll
        for (int t = 0; t < NT; ++t) {
            // B fragment: VGPR0 = B[k+2*hi][col], VGPR1 = B[k+2*hi+1][col]
            v2f b;
            b[0] = bcol[k * N + 16 * t];
            b[1] = bcol[k * N + 16 * t + N];
            acc[t] = __builtin_amdgcn_wmma_f32_16x16x4_f32(false, a, false, b,
                                                           (short)0, acc[t],
                                                           false, false);
        }
    }

#pragma unroll
    for (int t = 0; t < NT; ++t) {
        int col = n0 + 16 * t + l;
        float bb = BIAS ? bias[col] : 0.0f;
#pragma unroll
        for (int r = 0; r < 8; ++r) {
            float v = acc[t][r] + bb;
            if (RELU) v = fmaxf(v, 0.0f);
            C[(row0 + r + 8 * hi) * N + col] = v;
        }
    }
}

// 16 threads per edge, one float4 each: agg[dst] += norm * hw[src]
__global__ void edge_msg_kernel(const int* __restrict__ src, const int* __restrict__ dst,
                                const float* __restrict__ hw, const float* __restrict__ dinv,
                                float* __restrict__ agg, int E) {
    int t = blockIdx.x * blockDim.x + threadIdx.x;
    int e = t >> 4;
    if (e >= E) return;
    int part = t & 15;
    int s = src[e], d = dst[e];
    float norm = dinv[s] * dinv[d];
    float4 v = ((const float4*)(hw + s * DIM))[part];
    float* o = agg + d * DIM + part * 4;
    atomic_add_f32(o + 0, v.x * norm);
    atomic_add_f32(o + 1, v.y * norm);
    atomic_add_f32(o + 2, v.z * norm);
    atomic_add_f32(o + 3, v.w * norm);
}

// h_out = relu(agg + dinv^2 * hw + bias)
__global__ void finalize_kernel(const float* __restrict__ agg, const float* __restrict__ hw,
                                const float* __restrict__ dinv, const float* __restrict__ bias,
                                float* __restrict__ hout, int total) {
    int t = blockIdx.x * blockDim.x + threadIdx.x;
    if (t >= total) return;
    int i = t >> 6;
    int d = t & (DIM - 1);
    float di = dinv[i];
    float v = agg[t] + di * di * hw[t] + bias[d];
    hout[t] = fmaxf(v, 0.0f);
}

// g[batch[i]] += h[i]
__global__ void pool_kernel(const int* __restrict__ batch, const float* __restrict__ h,
                            float* __restrict__ g, int total) {
    int t = blockIdx.x * blockDim.x + threadIdx.x;
    if (t >= total) return;
    int i = t >> 6;
    int d = t & (DIM - 1);
    atomic_add_f32(&g[batch[i] * DIM + d], h[t]);
}

// out[i] = g1[i,:] . W + b   (K = 128, trivial)
__global__ void fc2_kernel(const float* __restrict__ g1, const float* __restrict__ W,
                           const float* __restrict__ b, float* __restrict__ out, int G) {
    int i = blockIdx.x * blockDim.x + threadIdx.x;
    if (i >= G) return;
    float s = b[0];
#pragma unroll 8
    for (int j = 0; j < FC1; ++j) s += g1[i * FC1 + j] * W[j];
    out[i] = s;
}

extern "C" void kernel_launch(void* const* d_in, const int* in_sizes, int n_in,
                              void* d_out, int out_size, void* d_ws, size_t ws_size,
                              hipStream_t stream) {
    const float* x     = (const float*)d_in[0];
    const int*   eidx  = (const int*)d_in[1];
    const int*   src   = eidx;               // edge_index[0]
    const int*   dst   = eidx + N_EDGES;     // edge_index[1]
    const int*   batch = (const int*)d_in[2];
    const float* W[5];
    const float* b[5];
    for (int i = 0; i < 5; ++i) {
        W[i] = (const float*)d_in[3 + 2 * i];
        b[i] = (const float*)d_in[4 + 2 * i];
    }
    const float* fc1W = (const float*)d_in[13];
    const float* fc1b = (const float*)d_in[14];
    const float* fc2W = (const float*)d_in[15];
    const float* fc2b = (const float*)d_in[16];
    float* out = (float*)d_out;

    const size_t NH = (size_t)N_NODES * DIM;     // 3.2M floats
    float* f    = (float*)d_ws;
    float* dinv = f;                              // 50048 (padded)
    float* hw   = f + 50048;                      // NH
    float* agg  = hw + NH;                        // NH
    float* h0   = agg + NH;                       // NH
    float* h1   = h0 + NH;                        // NH
    float* g    = h1 + NH;                        // 512*64
    float* g1   = g + NUM_GRAPHS * DIM;           // 512*128

    const int T = 256;

    // dinv = rsqrt(deg + 1)
    zero_kernel<<<(N_NODES + T - 1) / T, T, 0, stream>>>(dinv, N_NODES);
    degree_kernel<<<(N_EDGES + T - 1) / T, T, 0, stream>>>(dst, dinv, N_EDGES);
    rsqrt_kernel<<<(N_NODES + T - 1) / T, T, 0, stream>>>(dinv, N_NODES);

    const float* hin = x;
    const int nh = (int)NH;
    for (int layer = 0; layer < 5; ++layer) {
        float* hout = (layer & 1) ? h1 : h0;
        // hw = hin @ W[layer]: one wave per 16x64 strip -> 3125 waves
        int waves = N_NODES / 16;
        wmma_gemm<DIM, DIM, 4, false, false>
            <<<(waves + 3) / 4, 128, 0, stream>>>(hin, W[layer], nullptr, hw, N_NODES / 16);
        zero_kernel<<<(nh + T - 1) / T, T, 0, stream>>>(agg, nh);
        int et = N_EDGES * 16;                                 // 12.8M threads
        edge_msg_kernel<<<(et + T - 1) / T, T, 0, stream>>>(src, dst, hw, dinv, agg, N_EDGES);
        finalize_kernel<<<(nh + T - 1) / T, T, 0, stream>>>(agg, hw, dinv, b[layer], hout, nh);
        hin = hout;
    }

    // readout
    zero_kernel<<<(NUM_GRAPHS * DIM + T - 1) / T, T, 0, stream>>>(g, NUM_GRAPHS * DIM);
    pool_kernel<<<(nh + T - 1) / T, T, 0, stream>>>(batch, hin, g, nh);

    // fc1: [512x64] @ [64x128] + bias, relu: 32 row-tiles x 2 col-groups = 64 waves
    int waves1 = (NUM_GRAPHS / 16) * (FC1 / 64);
    wmma_gemm<DIM, FC1, 4, true, true>
        <<<(waves1 + 3) / 4, 128, 0, stream>>>(g, fc1W, fc1b, g1, NUM_GRAPHS / 16);
    // fc2: [512x128] @ [128x1] + bias
    fc2_kernel<<<(NUM_GRAPHS + T - 1) / T, T, 0, stream>>>(g1, fc2W, fc2b, out, NUM_GRAPHS);
}